// LocalGMMScorerAttention_46737834115652
// MI455X (gfx1250) — compile-verified
//
#include <hip/hip_runtime.h>

// Problem constants (match reference setup_inputs)
constexpr int Bn = 32;     // batch
constexpr int Ln = 2048;   // context length
constexpr int Dn = 512;    // ctx feature dim
constexpr int Qn = 512;    // query dim
constexpr int Hn = 256;    // hidden dim
constexpr int Kn = 4;      // mixture components
constexpr int MR = 32;     // padded candidate rows (>= K*7 = 28)
constexpr int STRIDE = Dn + 4;  // LDS row stride (516) -> bank-conflict-free

typedef __attribute__((ext_vector_type(2))) float v2f;
typedef __attribute__((ext_vector_type(8))) float v8f;

// ---------------------------------------------------------------------------
// Kernel 1: per-batch GMM stats (alpha/beta/kappa) + query projection qproj
// grid = B, block = H (256 threads)
// ws layout (floats): alpha[B*K] | beta[B*K] | kappa[B*K] | qproj[B*H]
// ---------------------------------------------------------------------------
__global__ __launch_bounds__(Hn) void gmm_stats_kernel(
    const float* __restrict__ query,   // (B, Q)
    const float* __restrict__ W_q2p,   // (Q, H)
    const float* __restrict__ b_q2p,   // (H)
    const float* __restrict__ W_p2s,   // (H, 3K)
    const float* __restrict__ b_p2s,   // (3K)
    const float* __restrict__ W_s0,    // (D+Q, H)
    float* __restrict__ ws) {
  __shared__ float h_sh[Hn];
  const int b = blockIdx.x;
  const int j = threadIdx.x;

  float* alpha = ws;
  float* beta  = ws + Bn * Kn;
  float* kappa = ws + 2 * Bn * Kn;
  float* qproj = ws + 3 * Bn * Kn;

  float accH = b_q2p[j];
  float accQ = 0.f;
  const float* qrow = query + (long)b * Qn;
  for (int q = 0; q < Qn; ++q) {
    const float qv = qrow[q];
    accH += qv * W_q2p[(long)q * Hn + j];
    accQ += qv * W_s0[(long)(Dn + q) * Hn + j];
  }
  h_sh[j] = tanhf(accH);
  qproj[(long)b * Hn + j] = accQ;
  __syncthreads();

  if (j < 3 * Kn) {
    float s = b_p2s[j];
    for (int i = 0; i < Hn; ++i) s += h_sh[i] * W_p2s[i * 3 * Kn + j];
    const float e = expf(s);
    if (j < Kn)              alpha[b * Kn + j] = e;
    else if (j < 2 * Kn)     beta[b * Kn + j - Kn] = e;
    else                     kappa[b * Kn + j - 2 * Kn] = e;
  }
}

// ---------------------------------------------------------------------------
// Kernel 2: pruned-window scorer (WMMA f32) + normalize + outputs
// grid = B, block = 256 (8 waves)
// out layout: expected_ctx (B,D) followed by p_ctx (B,L)
// ---------------------------------------------------------------------------
__global__ __launch_bounds__(256) void window_attend_kernel(
    const float* __restrict__ ctx,    // (B, L, D)
    const float* __restrict__ W_s0,   // (D+Q, H)
    const float* __restrict__ b_s0,   // (H)
    const float* __restrict__ W_s1,   // (H)
    const float* __restrict__ b_s1,   // (1)
    const float* __restrict__ ws,
    float* __restrict__ out) {
  __shared__ float comb[MR * STRIDE];   // gathered ctx rows (zero padded)
  __shared__ float scoreSh[MR];
  __shared__ float pSh[MR];
  __shared__ int   candSh[MR];
  __shared__ int   ncSh;
  __shared__ float aSh[Kn], bSh[Kn], kSh[Kn], cSh[Kn];
  __shared__ float invSumSh;

  const int b   = blockIdx.x;
  const int tid = threadIdx.x;

  const float* alpha = ws;
  const float* beta  = ws + Bn * Kn;
  const float* kappa = ws + 2 * Bn * Kn;
  const float* qproj = ws + 3 * Bn * Kn;

  float* outE = out;               // (B, D)
  float* outP = out + Bn * Dn;     // (B, L)

  if (tid < MR) scoreSh[tid] = 0.f;
  if (tid == 0) {
    int nc = 0;
    for (int k = 0; k < Kn; ++k) {
      const float ka = kappa[b * Kn + k];
      aSh[k] = alpha[b * Kn + k];
      bSh[k] = beta[b * Kn + k];
      kSh[k] = ka;
      const float cf = rintf(ka);   // round-half-even, matches jnp.round
      cSh[k] = cf;
      if (!(cf >= -4.f && cf <= (float)(Ln + 3))) continue;
      const int c = (int)cf;
      for (int off = -3; off <= 3; ++off) {
        const int p = c + off;
        if (p < 0 || p >= Ln) continue;
        bool dup = false;
        for (int r = 0; r < nc; ++r)
          if (candSh[r] == p) { dup = true; break; }
        if (!dup && nc < MR) candSh[nc++] = p;
      }
    }
    ncSh = nc;
  }
  __syncthreads();
  const int nc = ncSh;

  // zero-fill this batch's p_ctx row (mostly-zero output)
  for (int l = tid; l < Ln; l += blockDim.x) outP[(long)b * Ln + l] = 0.f;

  // stage gathered ctx rows into LDS (pad unused rows with zeros)
  for (int idx = tid; idx < MR * Dn; idx += blockDim.x) {
    const int r = idx / Dn;
    const int c = idx - r * Dn;
    comb[r * STRIDE + c] =
        (r < nc) ? ctx[((long)b * Ln + candSh[r]) * (long)Dn + c] : 0.f;
  }
  __syncthreads();

  // --- WMMA f32 16x16x4 GEMM: (MR x D) @ (D x H), 32 tiles over 8 waves ---
  const int wave = tid >> 5;
  const int lane = tid & 31;
  const int half = lane >> 4;   // 0: lanes 0-15, 1: lanes 16-31
  const int ln   = lane & 15;

  const float* bs1 = b_s1;   // scalar
  for (int tt = wave; tt < 32; tt += 8) {
    const int mt = tt >> 4;       // 0..1
    const int nt = tt & 15;       // 0..15
    const int col = nt * 16 + ln; // output column (0..255)
    const float* Bp = W_s0 + col;                       // stride H between k-rows
    const float* Ap = &comb[(mt * 16 + ln) * STRIDE + 2 * half];

    v8f acc = {};
    for (int kk = 0; kk < Dn; kk += 4) {
      v2f av, bv;
      av[0] = Ap[kk];
      av[1] = Ap[kk + 1];
      bv[0] = Bp[(long)(kk + 2 * half) * Hn];
      bv[1] = Bp[(long)(kk + 2 * half + 1) * Hn];
      acc = __builtin_amdgcn_wmma_f32_16x16x4_f32(
          false, av, false, bv, (short)0, acc, false, false);
    }

    const float qv = qproj[(long)b * Hn + col] + b_s0[col];
    const float wv = W_s1[col];
#pragma unroll
    for (int v = 0; v < 8; ++v) {
      const int row = mt * 16 + v + 8 * half;
      atomicAdd(&scoreSh[row], tanhf(acc[v] + qv) * wv);
    }
  }
  __syncthreads();

  // prior * likelihood at candidate positions
  if (tid < nc) {
    const float pos = (float)candSh[tid];
    float prior = 0.f;
    for (int k = 0; k < Kn; ++k) {
      if (pos >= cSh[k] - 3.f && pos <= cSh[k] + 3.f) {
        const float dd = pos - kSh[k];
        prior += aSh[k] * expf(-bSh[k] * dd * dd);
      }
    }
    pSh[tid] = prior * expf(scoreSh[tid] + bs1[0]);
  }
  __syncthreads();
  if (tid == 0) {
    float s = 0.f;
    for (int r = 0; r < nc; ++r) s += pSh[r];
    invSumSh = 1.f / s;
  }
  __syncthreads();
  const float inv = invSumSh;

  if (tid < nc) outP[(long)b * Ln + candSh[tid]] = pSh[tid] * inv;

  // expected_ctx from the LDS-resident gathered rows
  for (int d0 = tid; d0 < Dn; d0 += blockDim.x) {
    float s = 0.f;
    for (int r = 0; r < nc; ++r) s += pSh[r] * comb[r * STRIDE + d0];
    outE[(long)b * Dn + d0] = s * inv;
  }
}

// ---------------------------------------------------------------------------
extern "C" void kernel_launch(void* const* d_in, const int* in_sizes, int n_in,
                              void* d_out, int out_size, void* d_ws,
                              size_t ws_size, hipStream_t stream) {
  const float* ctx   = (const float*)d_in[0];
  const float* query = (const float*)d_in[1];
  const float* W_q2p = (const float*)d_in[2];
  const float* b_q2p = (const float*)d_in[3];
  const float* W_p2s = (const float*)d_in[4];
  const float* b_p2s = (const float*)d_in[5];
  const float* W_s0  = (const float*)d_in[6];
  const float* b_s0  = (const float*)d_in[7];
  const float* W_s1  = (const float*)d_in[8];
  const float* b_s1  = (const float*)d_in[9];
  float* out = (float*)d_out;
  float* ws  = (float*)d_ws;

  gmm_stats_kernel<<<Bn, Hn, 0, stream>>>(query, W_q2p, b_q2p, W_p2s, b_p2s,
                                          W_s0, ws);
  window_attend_kernel<<<Bn, 256, 0, stream>>>(ctx, W_s0, b_s0, W_s1, b_s1,
                                               ws, out);
}